// FPVectorField_29025388986482
// MI455X (gfx1250) — compile-verified
//
#include <hip/hip_runtime.h>
#include <cmath>

// Problem constants (match reference)
#define B_ROWS   16384
#define NVC      1024
#define VMAXF    6.0f
#define SC_ITERS 10

typedef __attribute__((ext_vector_type(2))) float v2f;
typedef __attribute__((ext_vector_type(8))) float v8f;

// ---------------------------------------------------------------------------
// Kernel 1a: moments via WMMA f32 16x16x4.
// One wave (32 threads) handles 16 rows. D = A(16x4 of f) x B(4x16 weights),
// accumulated over 256 K-chunks. B columns: n=0 -> 1, n=1 -> v, n=2 -> v^2,
// generated as a per-lane 2-FMA polynomial (c2*v + c1)*v + c0 with hoisted
// 0/1 coefficients (no per-iteration selects). Cell center v is an induction
// variable. A layout (ISA 7.12.2, 32-bit A 16x4): lanes 0-15 hold M=0..15 /
// K=0,1 in VGPR0,1 ; lanes 16-31 hold K=2,3. B mirrors (K=0,1 | K=2,3).
// ---------------------------------------------------------------------------
__global__ __launch_bounds__(32) void fp_moments_wmma(
    const float* __restrict__ f, const float* __restrict__ dv_p,
    float4* __restrict__ scal) {
  const int lane    = threadIdx.x;       // 0..31
  const int rowbase = blockIdx.x * 16;
  const float dv    = *dv_p;
  const int r       = lane & 15;         // A: row ; B: output column n
  const int khalf   = (lane >> 4) * 2;   // 0 for lanes 0-15, 2 for lanes 16-31

  const float* fp = f + (size_t)(rowbase + r) * NVC + khalf;

  // hoisted per-lane weight-polynomial coefficients: w(v) = c2*v^2 + c1*v + c0
  const float c0 = (r == 0) ? 1.0f : 0.0f;
  const float c1 = (r == 1) ? 1.0f : 0.0f;
  const float c2 = (r == 2) ? 1.0f : 0.0f;

  const float dv4 = 4.0f * dv;
  float vcur = dv * ((float)khalf + 0.5f) - VMAXF;  // v at k = khalf

  v8f acc0 = {}, acc1 = {}, acc2 = {}, acc3 = {};
  for (int kb = 0; kb < NVC; kb += 16) {
#pragma unroll
    for (int u = 0; u < 4; ++u) {
      const int k = kb + 4 * u;
      v2f a;
      a.x = fp[k + 0];
      a.y = fp[k + 1];
      const float v0 = vcur;
      const float v1 = vcur + dv;
      v2f b;
      b.x = (c2 * v0 + c1) * v0 + c0;   // 2 FMAs
      b.y = (c2 * v1 + c1) * v1 + c0;   // 2 FMAs
      v8f* acc = (u == 0) ? &acc0 : (u == 1) ? &acc1 : (u == 2) ? &acc2 : &acc3;
      *acc = __builtin_amdgcn_wmma_f32_16x16x4_f32(
          false, a, false, b, (short)0, *acc, false, false);
      vcur += dv4;
    }
  }
  v8f acc = (acc0 + acc1) + (acc2 + acc3);

  float av[8];
#pragma unroll
  for (int i = 0; i < 8; ++i) av[i] = acc[i];

  // C/D layout: D[m][n] lives in VGPR (m&7), lane n (m<8) or lane 16+n (m>=8)
  float myn = 0.f, myvb = 0.f, myet = 0.f, mybeta = 0.f;
#pragma unroll
  for (int m = 0; m < 16; ++m) {
    const int base  = (m < 8) ? 0 : 16;
    const float val = av[m & 7];
    const float S0  = __shfl(val, base + 0, 32);
    const float S1  = __shfl(val, base + 1, 32);
    const float S2  = __shfl(val, base + 2, 32);
    if (lane == m) {
      const float n_   = S0 * dv;
      const float vbar = S1 / S0;
      const float e_t  = (S2 - S1 * S1 / S0) * dv;  // sum f*(v-vbar)^2 * dv
      myn = n_; myvb = vbar; myet = e_t;
      mybeta = n_ / (2.0f * e_t);
    }
  }
  if (lane < 16) scal[rowbase + lane] = make_float4(myn, myvb, myet, mybeta);
}

// ---------------------------------------------------------------------------
// Kernel 1b: self-consistent beta loop. Wave per row, lane owns 32 cells.
// Entirely memory-free inside the loop (u2 analytic, in registers).
// ---------------------------------------------------------------------------
__global__ __launch_bounds__(256) void fp_beta_iter(
    const float* __restrict__ dv_p, float4* __restrict__ scal) {
  const int wave = threadIdx.x >> 5;
  const int lane = threadIdx.x & 31;
  const int row  = blockIdx.x * 8 + wave;
  const float dv = *dv_p;

  const float4 s = scal[row];
  const float n_ = s.x, vbar = s.y, e_t = s.z;
  float beta     = s.w;
  const float g  = n_ / e_t;

  float u2[32];
#pragma unroll
  for (int c = 0; c < 32; ++c) {
    const int j   = lane + 32 * c;
    const float v = dv * ((float)j + 0.5f) - VMAXF;
    const float u = v - vbar;
    u2[c] = u * u;
  }

  for (int it = 0; it < SC_ITERS; ++it) {
    float sm = 0.f, se = 0.f;
#pragma unroll
    for (int c = 0; c < 32; ++c) {
      const float M = __expf(-beta * u2[c]);
      sm += M;
      se += M * u2[c];
    }
#pragma unroll
    for (int off = 16; off > 0; off >>= 1) {
      sm += __shfl_xor(sm, off, 32);
      se += __shfl_xor(se, off, 32);
    }
    beta = beta * (se / sm) * g;   // beta *= e_d * n / e_t
  }
  if (lane == 0) scal[row].w = beta;
}

// ---------------------------------------------------------------------------
// Kernel 2: fused Chang-Cooper coefficients + Thomas solve. Thread per row.
// Since 2*beta*D == 1: C_edge = (v_edge - vbar). Edge state (Ce, P, delta) is
// carried j -> j+1 so each j computes exactly one new edge (1 expm1 + 1 div).
// cp stored TRANSPOSED in ws (coalesced); dp lives in d_out (float4, in-place
// backward sweep).
// ---------------------------------------------------------------------------
__global__ __launch_bounds__(256) void fp_thomas(
    const float* __restrict__ f, const float* __restrict__ v_edge,
    const float* __restrict__ dv_p, const float* __restrict__ nu_p,
    const float* __restrict__ dt_p, const float4* __restrict__ scal,
    float* __restrict__ cp_ws, float* __restrict__ out) {
  const int row  = blockIdx.x * blockDim.x + threadIdx.x;
  const float dv = *dv_p, nu = *nu_p, dt = *dt_p;

  const float4 sc  = scal[row];
  const float vbar = sc.y, beta = sc.w;
  const float D      = 0.5f / beta;
  const float Pdv    = D / dv;            // De/dv at interior edges
  const float s      = dt * nu / dv;
  const float twobdv = 2.0f * beta * dv;  // w = 2*beta*dv*(v_edge - vbar)

  const float* frow = f + (size_t)row * NVC;
  float*       orow = out + (size_t)row * NVC;

  // edge 0 is a zero-flux boundary
  float Ce0 = 0.f, P0 = 0.f, d0 = 0.5f;
  float cp_prev = 0.f, dp_prev = 0.f;

  for (int jb = 0; jb < NVC; jb += 4) {
    const float4 fj = *(const float4*)(frow + jb);
    float fv[4] = {fj.x, fj.y, fj.z, fj.w};
    float dpv[4];
#pragma unroll
    for (int q = 0; q < 4; ++q) {
      const int j = jb + q;
      const int e = j + 1;
      float Ce1, P1, d1;
      if (e < NVC) {  // interior edge
        const float c = v_edge[e] - vbar;  // uniform e -> scalar load of v_edge
        Ce1 = c;
        P1  = Pdv;
        const float w = twobdv * c;
        if (fabsf(w) < 1e-6f) {
          d1 = 0.5f;
        } else {
          const float em = expm1f(w);
          d1 = (em - w) / (w * em);        // 1/w - 1/expm1(w), single divide
        }
      } else {        // edge NV: zero-flux boundary
        Ce1 = 0.f; P1 = 0.f; d1 = 0.5f;
      }
      const float coefU = Ce1 * (1.0f - d1) + P1;
      const float coefL = -Ce0 * d0 + P0;
      const float coefD = Ce1 * d1 - P1 - Ce0 * (1.0f - d0) - P0;
      const float dl = -s * coefL;
      const float dd = 1.0f - s * coefD;
      const float du = -s * coefU;

      const float denom = dd - dl * cp_prev;
      const float rdiv  = 1.0f / denom;
      const float cp    = du * rdiv;
      const float dp    = (fv[q] - dl * dp_prev) * rdiv;

      cp_ws[(size_t)j * B_ROWS + row] = cp;  // transposed -> coalesced
      dpv[q]  = dp;
      cp_prev = cp; dp_prev = dp;
      Ce0 = Ce1; P0 = P1; d0 = d1;
    }
    *(float4*)(orow + jb) = make_float4(dpv[0], dpv[1], dpv[2], dpv[3]);
  }

  // backward sweep, in place over d_out
  float xnext = 0.f;
  for (int jb = NVC - 4; jb >= 0; jb -= 4) {
    const float4 dpj = *(const float4*)(orow + jb);
    float dv4[4] = {dpj.x, dpj.y, dpj.z, dpj.w};
    float xv[4];
#pragma unroll
    for (int q = 3; q >= 0; --q) {
      const int j   = jb + q;
      const float c = cp_ws[(size_t)j * B_ROWS + row];
      const float x = dv4[q] - c * xnext;
      xv[q] = x;
      xnext = x;
    }
    *(float4*)(orow + jb) = make_float4(xv[0], xv[1], xv[2], xv[3]);
  }
}

// ---------------------------------------------------------------------------
extern "C" void kernel_launch(void* const* d_in, const int* in_sizes, int n_in,
                              void* d_out, int out_size, void* d_ws, size_t ws_size,
                              hipStream_t stream) {
  const float* f      = (const float*)d_in[0];
  const float* v      = (const float*)d_in[1];  (void)v;  // computed analytically
  const float* v_edge = (const float*)d_in[2];
  const float* dv     = (const float*)d_in[3];
  const float* nu     = (const float*)d_in[4];
  const float* dt     = (const float*)d_in[5];
  (void)in_sizes; (void)n_in; (void)out_size; (void)ws_size;

  // ws layout: [16384 x float4 row scalars (n, vbar, e_t, beta)] [cp: B*NV f32]
  float4* scal = (float4*)d_ws;
  float*  cp   = (float*)((char*)d_ws + (size_t)B_ROWS * sizeof(float4));
  float*  out  = (float*)d_out;

  fp_moments_wmma<<<B_ROWS / 16, 32, 0, stream>>>(f, dv, scal);
  fp_beta_iter  <<<B_ROWS / 8, 256, 0, stream>>>(dv, scal);
  fp_thomas     <<<B_ROWS / 256, 256, 0, stream>>>(f, v_edge, dv, nu, dt, scal, cp, out);
}